// DefaultProjectorTorch_65206193488358
// MI455X (gfx1250) — compile-verified
//
#include <hip/hip_runtime.h>
#include <hip/hip_bf16.h>
#include <math.h>

// ---------------- problem constants (mirror the reference) ----------------
#define GRID   256
#define DDIM   41                  // 2*RMAX+1, RMAX=20
#define NPTS   (DDIM*DDIM*DDIM)    // 68921
#define NATOMS 32
#define BLOCK  256                 // 8 wave32 waves
#define NWAVES 8
#define SPLIT  32                  // blocks per atom  -> 1024 blocks total
#define CHUNK  ((NPTS + SPLIT - 1) / SPLIT)          // 2154 points per block
#define NITERS ((CHUNK + BLOCK - 1) / BLOCK)         // 9 iterations
#define ACELL  0.1f                // 25.6/256
#define VCELL  1.0e-3f
#define R_O    2.0f

typedef float v2f __attribute__((ext_vector_type(2)));
typedef float v8f __attribute__((ext_vector_type(8)));

// Wave-local LDS ordering: staging tiles are wave-private and CDNA5 LDS ops
// from one wave are executed in-order (DScnt, ISA 7.3). We only need to stop
// the COMPILER from reordering LDS ops across this point; no s_barrier needed.
#define WAVE_FENCE() do { asm volatile("" ::: "memory"); __builtin_amdgcn_wave_barrier(); } while (0)

__global__ __launch_bounds__(BLOCK)
void soap_project_wmma(const float* __restrict__ rho,
                       const float* __restrict__ positions,
                       const float* __restrict__ Wmat,
                       const int*   __restrict__ offs,
                       float* __restrict__ out)
{
    // ---------------- LDS ----------------
    __shared__ int   sIdx[3][DDIM];                         // wrapped grid indices per axis
    __shared__ float sCrd[3][DDIM];                         // A*offs - dr per axis
    __shared__ __align__(16) float sAng[NWAVES][32 * 16];   // A staging (angulars)
    __shared__ __align__(16) float sB  [NWAVES][32 * 16];   // B staging (cols 4..15 = 0)
    __shared__ float sPart[NWAVES][64];                     // per-wave partial C

    const int tid   = threadIdx.x;
    const int lid   = tid & 31;               // lane in wave (wave32)
    const int wv    = tid >> 5;               // wave id in block
    const int atom  = blockIdx.x / SPLIT;
    const int split = blockIdx.x % SPLIT;
    const int pBase = split * CHUNK;
    const int pEnd  = (pBase + CHUNK < NPTS) ? (pBase + CHUNK) : NPTS;

    // ---------------- per-atom setup (uniform per block) ----------------
    const float px = positions[atom * 3 + 0];
    const float py = positions[atom * 3 + 1];
    const float pz = positions[atom * 3 + 2];
    const float cmx = rintf(px / ACELL), cmy = rintf(py / ACELL), cmz = rintf(pz / ACELL);
    const float drx = px - ACELL * cmx, dry = py - ACELL * cmy, drz = pz - ACELL * cmz;
    const int   cmi[3] = { (int)cmx, (int)cmy, (int)cmz };
    const float drv[3] = { drx, dry, drz };

    if (tid < 3 * DDIM) {
        const int axis = tid / DDIM;
        const int i    = tid - axis * DDIM;
        const int o    = offs[i];
        sIdx[axis][i]  = (o + cmi[axis]) & (GRID - 1);      // mod 256 (handles negatives)
        sCrd[axis][i]  = ACELL * (float)o - drv[axis];
    }

    float* const angRow = &sAng[wv][lid * 16];
    float* const bRow   = &sB  [wv][lid * 16];

    // zero this lane's own B row once: columns n=4..15 then stay zero forever
    {
        const float4 z4 = make_float4(0.f, 0.f, 0.f, 0.f);
        *(float4*)(bRow +  0) = z4;
        *(float4*)(bRow +  4) = z4;
        *(float4*)(bRow +  8) = z4;
        *(float4*)(bRow + 12) = z4;
    }

    // W (4x4, row-major [n][m]) -> scalar regs (uniform)
    float Wr[16];
#pragma unroll
    for (int i = 0; i < 16; ++i) Wr[i] = Wmat[i];

    // radial normalizations (compile-time folded): Nn[a] = sqrt(720*2^(11+2a)/prod)
    const float iN0 = 1.0f / sqrtf(1474560.0f  / 1663200.0f);
    const float iN1 = 1.0f / sqrtf(5898240.0f  / 8648640.0f);
    const float iN2 = 1.0f / sqrtf(23592960.0f / 32432400.0f);
    const float iN3 = 1.0f / sqrtf(94371840.0f / 98017920.0f);

    // Cartesian real-spherical-harmonic coefficients (compiler-folded literals)
    const float SQ2 = 1.4142135623730951f;
    const float C00 = 0.28209479177387814f;                    // Y00
    const float C1  = SQ2 * 0.34549414947133550f;              // * y , * x
    const float C10 = 0.48860251190291992f;                    // * z
    const float C22 = 6.0f  * SQ2 * 0.12875806734109888f;      // * xy
    const float C21 = 3.0f  * SQ2 * 0.25751613468219770f;      // * zy, * zx
    const float C20 = 0.63078313050504010f;                    // * (1.5 z^2 - 0.5)
    const float C2c = 3.0f  * SQ2 * 0.12875806734109888f;      // * (x^2 - y^2)
    const float C33 = 15.0f * SQ2 * 0.027814921575518937f;     // * y(3x^2-y^2), x(x^2-3y^2)
    const float C32 = 30.0f * SQ2 * 0.06813236509555214f;      // * xyz
    const float C31 = 1.5f  * SQ2 * 0.21545530f;               // * y(5z^2-1), x(5z^2-1)
    const float C30 = 0.74635266518023080f;                    // * 0.5 z(5z^2-3)
    const float C3c = 15.0f * SQ2 * 0.06813236509555214f;      // * z(x^2-y^2)

    v8f acc = {0.f, 0.f, 0.f, 0.f, 0.f, 0.f, 0.f, 0.f};

    __syncthreads();   // axis tables visible to all waves

    for (int it = 0; it < NITERS; ++it) {
        const int p = pBase + it * BLOCK + tid;

        float ang[16];
        float b0 = 0.f, b1 = 0.f, b2 = 0.f, b3 = 0.f;
#pragma unroll
        for (int i = 0; i < 16; ++i) ang[i] = 0.f;

        if (p < pEnd) {
            const int ix = p / (DDIM * DDIM);
            const int r0 = p - ix * (DDIM * DDIM);
            const int iy = r0 / DDIM;
            const int iz = r0 - iy * DDIM;

            const float X = sCrd[0][ix], Y = sCrd[1][iy], Z = sCrd[2][iz];
            const float R2 = X * X + Y * Y + Z * Z;

            // Outside the cutoff sphere all radials are zero -> contribution is
            // zero regardless of angulars: skip the gather and ALL the math.
            if (R2 < R_O * R_O) {
                const float srho = rho[(sIdx[0][ix] << 16) | (sIdx[1][iy] << 8) | sIdx[2][iz]];

                const bool  small = (R2 < 1e-30f);
                const float invR  = small ? 0.0f : rsqrtf(R2);
                const float R     = R2 * invR;                       // 0 when small
                const float x     = X * invR;                        // st*cos(phi)
                const float y     = Y * invR;                        // st*sin(phi)
                float z           = small ? 1.0f : Z * invR;         // ct
                z = fminf(1.0f, fmaxf(-1.0f, z));

                const float x2 = x * x, y2 = y * y, z2 = z * z;

                ang[0]  = C00;
                ang[1]  = C1  * y;
                ang[2]  = C10 * z;
                ang[3]  = C1  * x;
                ang[4]  = C22 * x * y;
                ang[5]  = C21 * z * y;
                ang[6]  = C20 * (1.5f * z2 - 0.5f);
                ang[7]  = C21 * z * x;
                ang[8]  = C2c * (x2 - y2);
                ang[9]  = C33 * y * (3.f * x2 - y2);
                ang[10] = C32 * x * y * z;
                ang[11] = C31 * y * (5.f * z2 - 1.f);
                ang[12] = C30 * 0.5f * z * (5.f * z2 - 3.f);
                ang[13] = C31 * x * (5.f * z2 - 1.f);
                ang[14] = C3c * z * (x2 - y2);
                ang[15] = C33 * x * (x2 - 3.f * y2);

                const float t  = R_O - R;
                const float t2 = t * t;
                const float g0 = R2 * t2 * iN0;
                const float g1 = R2 * t2 * t * iN1;
                const float g2 = R2 * t2 * t2 * iN2;
                const float g3 = R2 * t2 * t2 * t * iN3;
                const float sc = VCELL * srho;
                b0 = sc * (Wr[0]  * g0 + Wr[1]  * g1 + Wr[2]  * g2 + Wr[3]  * g3);
                b1 = sc * (Wr[4]  * g0 + Wr[5]  * g1 + Wr[6]  * g2 + Wr[7]  * g3);
                b2 = sc * (Wr[8]  * g0 + Wr[9]  * g1 + Wr[10] * g2 + Wr[11] * g3);
                b3 = sc * (Wr[12] * g0 + Wr[13] * g1 + Wr[14] * g2 + Wr[15] * g3);
            }
        }

        // stage A (16 angulars) and B (4 radial*rho; cols 4..15 stay zero)
#pragma unroll
        for (int j = 0; j < 4; ++j) {
            float4 v = make_float4(ang[4 * j], ang[4 * j + 1], ang[4 * j + 2], ang[4 * j + 3]);
            *(float4*)(angRow + 4 * j) = v;
        }
        *(float4*)(bRow) = make_float4(b0, b1, b2, b3);

        WAVE_FENCE();   // stores -> loads (wave-private tile, HW DS in-order)

        // 8 x V_WMMA_F32_16X16X4_F32 consume this wave's 32 points (K = 4 each)
        const float* angBase = &sAng[wv][0];
        const float* bBase   = &sB[wv][0];
        const int half = lid >> 4;     // 0 | 1
        const int mn   = lid & 15;     // M for A, N for B
#pragma unroll
        for (int g = 0; g < 8; ++g) {
            const int pa = 4 * g + 2 * half;   // A: K = 0/1 (lanes 0-15), 2/3 (lanes 16-31)
            const int pb = 4 * g + half;       // B: rows K striped 2 per VGPR
            v2f a, b;
            a.x = angBase[pa * 16 + mn];
            a.y = angBase[(pa + 1) * 16 + mn];
            b.x = bBase[pb * 16 + mn];
            b.y = bBase[(pb + 2) * 16 + mn];
            acc = __builtin_amdgcn_wmma_f32_16x16x4_f32(
                false, a, false, b, (short)0, acc, false, false);
        }

        WAVE_FENCE();   // loads -> next iteration's stores
    }

    // ------------- extract C(16x4) and reduce across waves -------------
    // C layout: VGPR r, lanes 0-15 -> (M=r, N=lane); lanes 16-31 -> (M=r+8, N=lane-16)
    if ((lid & 15) < 4) {
        const int n     = lid & 15;
        const int baseM = (lid < 16) ? 0 : 8;
#pragma unroll
        for (int r = 0; r < 8; ++r)
            sPart[wv][n * 16 + baseM + r] = acc[r];
    }
    __syncthreads();

    if (tid < 64) {
        float s = 0.f;
#pragma unroll
        for (int w = 0; w < NWAVES; ++w) s += sPart[w][tid];
        atomicAdd(&out[atom * 64 + tid], s);   // out[n*16 + c] == coeff.reshape(-1)
    }
}

extern "C" void kernel_launch(void* const* d_in, const int* in_sizes, int n_in,
                              void* d_out, int out_size, void* d_ws, size_t ws_size,
                              hipStream_t stream) {
    const float* rho       = (const float*)d_in[0];
    const float* positions = (const float*)d_in[1];
    const float* Wmat      = (const float*)d_in[2];
    const int*   offs      = (const int*)d_in[3];
    float*       out       = (float*)d_out;

    // out accumulates via atomics -> must start at zero (harness poisons it)
    hipMemsetAsync(out, 0, (size_t)out_size * sizeof(float), stream);

    soap_project_wmma<<<NATOMS * SPLIT, BLOCK, 0, stream>>>(rho, positions, Wmat, offs, out);
}